// ODEIntegrator_57114475102837
// MI455X (gfx1250) — compile-verified
//
#include <hip/hip_runtime.h>

// ---------------------------------------------------------------------------
// Graph-SIR ODE integrator for MI455X (gfx1250).
//
// Math restructuring: x_in = [node_feat, I] =>
//   proj_msg  = node_feat@Wm_top + I (.) wm_last      (rank-1 time-varying part)
//   agg[d]    = AggP_static[d] + force_raw[d]*wm_last
//   h         = relu(Base[n] + force_raw[n]*wm_last + I[n]*ws_last)
//   Base      = AggP_static + node_feat@Ws_top + b_h        (time-invariant)
// so the 64-wide edge scatter (102M atomics) runs ONCE, and each of the 20
// steps costs only a scalar edge scatter + an L2-resident per-node pass.
// ---------------------------------------------------------------------------

typedef float v2f __attribute__((ext_vector_type(2)));
typedef float v8f __attribute__((ext_vector_type(8)));

#define DT_STEP 0.5f

__device__ __forceinline__ float sigmoidf_(float x) {
    return 1.0f / (1.0f + __expf(-x));
}

// ---- init SIR state + zero force accumulator ------------------------------
__global__ void k_init(const float* __restrict__ S0, const float* __restrict__ I0,
                       const float* __restrict__ R0,
                       float* __restrict__ S, float* __restrict__ I,
                       float* __restrict__ R, float* __restrict__ force, int N) {
    int n = blockIdx.x * blockDim.x + threadIdx.x;
    if (n < N) {
        S[n] = S0[n]; I[n] = I0[n]; R[n] = R0[n]; force[n] = 0.0f;
    }
}

// ---- WMMA fp32 GEMM: Pm = feat@Wm_top (row-major [N,64]),
// ----                 Base = feat@Ws_top + b_h (transposed [64,N]) ----------
__global__ void k_gemm(const float* __restrict__ feat,   // [N,64]
                       const float* __restrict__ Wmsg,   // [65,64]
                       const float* __restrict__ Wself,  // [65,64]
                       const float* __restrict__ bh,     // [64]
                       float* __restrict__ Pm,           // [N,64]
                       float* __restrict__ Base,         // [64,N]
                       int N) {
    const int lane = threadIdx.x & 31;
    const int wave = threadIdx.x >> 5;
    const int tile = blockIdx.x * (blockDim.x >> 5) + wave;
    const int m0   = tile * 16;
    if (m0 >= N) return;                      // uniform per wave -> EXEC stays full
    const int half = lane >> 4;               // 0: lanes 0-15, 1: lanes 16-31
    const int mr   = lane & 15;
    int row = m0 + mr; if (row >= N) row = N - 1;   // clamped, branch-free

    v8f accM[4], accS[4];
    #pragma unroll
    for (int nt = 0; nt < 4; ++nt)
        #pragma unroll
        for (int j = 0; j < 8; ++j) { accM[nt][j] = 0.0f; accS[nt][j] = 0.0f; }

    #pragma unroll
    for (int kk = 0; kk < 64; kk += 4) {
        // A fragment (16x4 f32): lanes 0-15 hold K=kk,kk+1; lanes 16-31 K=kk+2,kk+3
        const int kb = kk + half * 2;
        v2f a;
        a.x = feat[row * 64 + kb];
        a.y = feat[row * 64 + kb + 1];
        #pragma unroll
        for (int nt = 0; nt < 4; ++nt) {
            const int col = nt * 16 + mr;
            // B fragment (4x16 f32): VGPR0 rows K=kb, VGPR1 rows K=kb+1
            v2f bm, bs;
            bm.x = Wmsg[kb * 64 + col];
            bm.y = Wmsg[(kb + 1) * 64 + col];
            bs.x = Wself[kb * 64 + col];
            bs.y = Wself[(kb + 1) * 64 + col];
            accM[nt] = __builtin_amdgcn_wmma_f32_16x16x4_f32(
                false, a, false, bm, (short)0, accM[nt], false, false);
            accS[nt] = __builtin_amdgcn_wmma_f32_16x16x4_f32(
                false, a, false, bs, (short)0, accS[nt], false, false);
        }
    }
    // D fragment: VGPR v holds M = m0 + v + 8*half, N = col
    #pragma unroll
    for (int nt = 0; nt < 4; ++nt) {
        const int col = nt * 16 + mr;
        const float bhc = bh[col];
        #pragma unroll
        for (int v = 0; v < 8; ++v) {
            const int m = m0 + v + 8 * half;
            if (m < N) {
                Pm[(size_t)m * 64 + col]    = accM[nt][v];
                Base[(size_t)col * N + m]   = accS[nt][v] + bhc;
            }
        }
    }
}

// ---- one-time 64-wide edge scatter: Base[k][d] += w_e * Pm[src_e][k] -------
__global__ void k_agg(const int* __restrict__ ei, const float* __restrict__ ew,
                      const float* __restrict__ Pm, float* __restrict__ Base,
                      int N, int E) {
    long long tid = (long long)blockIdx.x * blockDim.x + threadIdx.x;
    int e  = (int)(tid >> 4);
    int kq = (int)(tid & 15);
    if (e < E) {
        int   s = ei[e];
        int   d = ei[E + e];
        float w = ew[e];
        const float4 p = *(const float4*)(Pm + (size_t)s * 64 + kq * 4);
        int k = kq * 4;
        unsafeAtomicAdd(&Base[(size_t)(k + 0) * N + d], w * p.x);
        unsafeAtomicAdd(&Base[(size_t)(k + 1) * N + d], w * p.y);
        unsafeAtomicAdd(&Base[(size_t)(k + 2) * N + d], w * p.z);
        unsafeAtomicAdd(&Base[(size_t)(k + 3) * N + d], w * p.w);
    }
}

// ---- per-step scalar edge scatter: force[d] += w_e * I[src_e] --------------
__global__ void k_force(const int* __restrict__ ei, const float* __restrict__ ew,
                        const float* __restrict__ I, float* __restrict__ force, int E) {
    int e = blockIdx.x * blockDim.x + threadIdx.x;
    if (e < E) {
        unsafeAtomicAdd(&force[ei[E + e]], ew[e] * I[ei[e]]);
    }
}

// ---- per-step node update (L2-resident streaming over Base) ----------------
__global__ void k_node(float* __restrict__ S, float* __restrict__ I, float* __restrict__ R,
                       float* __restrict__ force,
                       const float* __restrict__ Base,   // [64,N]
                       const float* __restrict__ Wmsg,   // row 64 = wm_last
                       const float* __restrict__ Wself,  // row 64 = ws_last
                       const float* __restrict__ wbeta, const float* __restrict__ bbeta,
                       const float* __restrict__ wgamma, const float* __restrict__ bgamma,
                       float* __restrict__ Iseq_t,
                       float* __restrict__ betaOut, float* __restrict__ gammaOut,
                       int N) {
    __shared__ float sWm[64], sWs[64], sWb[64], sWg[64];
    if (threadIdx.x < 64) {
        int k = threadIdx.x;
        sWm[k] = Wmsg[64 * 64 + k];
        sWs[k] = Wself[64 * 64 + k];
        sWb[k] = wbeta[k];
        sWg[k] = wgamma[k];
    }
    __syncthreads();
    int n = blockIdx.x * blockDim.x + threadIdx.x;
    if (n >= N) return;

    float f = force[n];
    force[n] = 0.0f;                 // self-reset for next step
    float In = I[n];
    float db = 0.0f, dg = 0.0f;
    #pragma unroll 8
    for (int k = 0; k < 64; ++k) {
        float h = Base[(size_t)k * N + n] + f * sWm[k] + In * sWs[k];
        h  = fmaxf(h, 0.0f);
        db = fmaf(h, sWb[k], db);
        dg = fmaf(h, sWg[k], dg);
    }
    float beta  = sigmoidf_(db + bbeta[0]);
    float gamma = sigmoidf_(dg + bgamma[0]);
    float fc  = fminf(fmaxf(f, 0.0f), 1000.0f);
    float Sn  = S[n], Rn = R[n];
    float inf = beta * Sn * fc;
    float rec = gamma * In;
    Sn = fminf(fmaxf(Sn - inf * DT_STEP, 0.0f), 1.0f);
    float In2 = fminf(fmaxf(In + (inf - rec) * DT_STEP, 0.0f), 1.0f);
    Rn = fminf(fmaxf(Rn + rec * DT_STEP, 0.0f), 1.0f);
    S[n] = Sn; I[n] = In2; R[n] = Rn;
    Iseq_t[n] = In2;
    if (betaOut) { betaOut[n] = beta; gammaOut[n] = gamma; }
}

extern "C" void kernel_launch(void* const* d_in, const int* in_sizes, int n_in,
                              void* d_out, int out_size, void* d_ws, size_t ws_size,
                              hipStream_t stream) {
    const float* S0    = (const float*)d_in[0];
    const float* I0    = (const float*)d_in[1];
    const float* R0    = (const float*)d_in[2];
    const int    T     = in_sizes[3];
    const float* feat  = (const float*)d_in[4];
    const int*   ei    = (const int*)d_in[5];   // [2,E] int32 (JAX x64 disabled)
    const float* ew    = (const float*)d_in[6];
    const float* Wmsg  = (const float*)d_in[7];
    const float* Wself = (const float*)d_in[8];
    const float* bh    = (const float*)d_in[9];
    const float* wbeta = (const float*)d_in[10];
    const float* bbeta = (const float*)d_in[11];
    const float* wgamma= (const float*)d_in[12];
    const float* bgamma= (const float*)d_in[13];
    const int N = in_sizes[0];
    const int E = in_sizes[6];

    // workspace: S,I,R,force (4N) + Pm (64N) + Base (64N)  ~= 52.8 MB
    float* S     = (float*)d_ws;
    float* I     = S + N;
    float* R     = I + N;
    float* force = R + N;
    float* Pm    = force + N;
    float* Base  = Pm + (size_t)N * 64;

    float* out      = (float*)d_out;
    float* Iseq     = out;
    float* betaOut  = out + (size_t)T * N;
    float* gammaOut = betaOut + N;

    k_init<<<(N + 255) / 256, 256, 0, stream>>>(S0, I0, R0, S, I, R, force, N);

    const int tiles = (N + 15) / 16;
    k_gemm<<<(tiles + 3) / 4, 128, 0, stream>>>(feat, Wmsg, Wself, bh, Pm, Base, N);

    const long long aggThreads = (long long)E * 16;
    k_agg<<<(unsigned)((aggThreads + 255) / 256), 256, 0, stream>>>(ei, ew, Pm, Base, N, E);

    for (int t = 0; t < T; ++t) {
        k_force<<<(E + 255) / 256, 256, 0, stream>>>(ei, ew, I, force, E);
        const bool last = (t == T - 1);
        k_node<<<(N + 255) / 256, 256, 0, stream>>>(
            S, I, R, force, Base, Wmsg, Wself,
            wbeta, bbeta, wgamma, bgamma,
            Iseq + (size_t)t * N,
            last ? betaOut : nullptr, last ? gammaOut : nullptr, N);
    }
}